// GNN_4595615007018
// MI455X (gfx1250) — compile-verified
//
#include <hip/hip_runtime.h>

typedef __attribute__((ext_vector_type(2))) float v2f;
typedef __attribute__((ext_vector_type(8))) float v8f;

#define HID 128

// ---------------- init: deg=1, agg=0, out=0 ----------------
__global__ void k_init(float* __restrict__ deg, float* __restrict__ agg,
                       float* __restrict__ out, int N) {
  long long idx = (long long)blockIdx.x * blockDim.x + threadIdx.x;
  long long tot = (long long)N * HID;
  if (idx < tot) agg[idx] = 0.0f;
  if (idx < N)   deg[idx] = 1.0f;   // self-loop
  if (idx < HID) out[idx] = 0.0f;
}

// ---------------- degree accumulation over dst ----------------
__global__ void k_deg(const int* __restrict__ dst, float* __restrict__ deg, int E) {
  int e = blockIdx.x * blockDim.x + threadIdx.x;
  if (e < E) atomicAdd(&deg[dst[e]], 1.0f);
}

__global__ void k_rsqrt(float* __restrict__ deg, int N) {
  int i = blockIdx.x * blockDim.x + threadIdx.x;
  if (i < N) deg[i] = rsqrtf(deg[i]);
}

// ---------------- Hs = rowscale(dinv) * (X @ W), WMMA f32 16x16x4 ----------------
// One wave -> 16 rows x 128 cols. Block = 8 waves = 128 rows. W staged in LDS (64KB).
// Safe to run in place (Hs == X): each A row is read only by its owning wave and
// all K-loop loads complete before the epilogue stores.
__global__ void __launch_bounds__(256) k_gemm_scale(const float* X,
                                                    const float* __restrict__ W,
                                                    const float* __restrict__ dinv,
                                                    float* Hs, int N) {
  __shared__ float lw[HID * HID];
  {
    const float4* Wv = (const float4*)W;
    float4* lv = (float4*)lw;
    for (int i = threadIdx.x; i < HID * HID / 4; i += 256) lv[i] = Wv[i];
  }
  __syncthreads();

  const int wave = threadIdx.x >> 5;
  const int lane = threadIdx.x & 31;
  const int tileM = blockIdx.x * 8 + wave;
  if (tileM * 16 >= N) return;                 // uniform per wave
  const int m0    = tileM * 16;
  const int laneM = lane & 15;                 // row within tile (A), col within ntile (B/D)
  const int koff  = (lane >> 4) << 1;          // lanes 16-31 hold K+2,K+3
  const float* xrow = X + (long long)(m0 + laneM) * HID;

  v8f acc[8];
#pragma unroll
  for (int t = 0; t < 8; ++t) acc[t] = (v8f){0.f,0.f,0.f,0.f,0.f,0.f,0.f,0.f};

  for (int k0 = 0; k0 < HID; k0 += 4) {
    v2f a;
    a.x = xrow[k0 + koff];
    a.y = xrow[k0 + koff + 1];
#pragma unroll
    for (int t = 0; t < 8; ++t) {
      const int n = t * 16 + laneM;
      v2f b;
      b.x = lw[(k0 + koff) * HID + n];
      b.y = lw[(k0 + koff + 1) * HID + n];
      acc[t] = __builtin_amdgcn_wmma_f32_16x16x4_f32(
          /*neg_a=*/false, a, /*neg_b=*/false, b,
          /*c_mod=*/(short)0, acc[t], /*reuse_a=*/false, /*reuse_b=*/false);
    }
  }

  // D layout: VGPR r -> (M=r, N=lane) lanes 0-15 ; (M=r+8, N=lane-16) lanes 16-31
  const int rbase = m0 + ((lane >> 4) << 3);
#pragma unroll
  for (int r = 0; r < 8; ++r) {
    const int row = rbase + r;
    const float s = dinv[row];
    float* orow = Hs + (long long)row * HID + laneM;
#pragma unroll
    for (int t = 0; t < 8; ++t) orow[t * 16] = acc[t][r] * s;
  }
}

// ---------------- edge scatter-add: one wave per edge, 512B row ----------------
__global__ void k_edge(const int* __restrict__ src, const int* __restrict__ dst,
                       const float* __restrict__ Hs, float* __restrict__ agg, int E) {
  long long tid = (long long)blockIdx.x * blockDim.x + threadIdx.x;
  int e = (int)(tid >> 5);
  if (e >= E) return;
  int l = (int)(tid & 31) * 4;
  int s = src[e];
  int d = dst[e];
  const float4 v = *(const float4*)(Hs + (long long)s * HID + l);
  float* p = agg + (long long)d * HID + l;
  atomicAdd(p + 0, v.x);
  atomicAdd(p + 1, v.y);
  atomicAdd(p + 2, v.z);
  atomicAdd(p + 3, v.w);
}

// ---------------- layer-1 epilogue: x2 = relu(dinv*(agg + Hs) + b); agg := 0 ----------------
__global__ void k_post_relu(float* __restrict__ A /* Hs in, x2 out */,
                            float* __restrict__ B /* agg in, zeroed out */,
                            const float* __restrict__ dinv,
                            const float* __restrict__ bias, int N) {
  long long idx = (long long)blockIdx.x * blockDim.x + threadIdx.x;
  if (idx >= (long long)N * HID) return;
  int i = (int)(idx >> 7);
  int c = (int)(idx & (HID - 1));
  float v = (B[idx] + A[idx]) * dinv[i] + bias[c];
  A[idx] = fmaxf(v, 0.0f);
  B[idx] = 0.0f;   // ready for layer-2 aggregation
}

// ---------------- final: out[c] = mean_i( dinv[i]*(agg+Hs) ) + b2[c] ----------------
__global__ void k_final(const float* __restrict__ A, const float* __restrict__ B,
                        const float* __restrict__ dinv, const float* __restrict__ bias,
                        float* __restrict__ out, int N) {
  const int c = threadIdx.x;               // 128 threads
  float local = 0.0f;
  for (int i = blockIdx.x; i < N; i += gridDim.x)
    local += (B[(long long)i * HID + c] + A[(long long)i * HID + c]) * dinv[i];
  atomicAdd(&out[c], local * (1.0f / (float)N));
  if (blockIdx.x == 0) atomicAdd(&out[c], bias[c]);
}

extern "C" void kernel_launch(void* const* d_in, const int* in_sizes, int n_in,
                              void* d_out, int out_size, void* d_ws, size_t ws_size,
                              hipStream_t stream) {
  const float* x  = (const float*)d_in[0];
  const int*   ei = (const int*)d_in[1];        // [2, E] flat: row0=src, row1=dst
  const float* W1 = (const float*)d_in[2];
  const float* b1 = (const float*)d_in[3];
  const float* W2 = (const float*)d_in[4];
  const float* b2 = (const float*)d_in[5];
  float* out = (float*)d_out;

  const int N = in_sizes[0] / HID;
  const int E = in_sizes[1] / 2;
  const int* src = ei;
  const int* dst = ei + E;

  float* ws   = (float*)d_ws;
  float* deg  = ws;                                   // N floats (becomes dinv)
  float* bufA = ws + (((long long)N + 3) & ~3LL);     // N*128: Hs, then x2, then Hs2
  float* bufB = bufA + (long long)N * HID;            // N*128: scatter accumulator

  const long long tot = (long long)N * HID;
  const int nhBlocks = (int)((tot + 255) / 256);
  const int gBlocks  = ((N + 15) / 16 + 7) / 8;
  const long long etot = (long long)E * 32;
  const int eBlocks  = (int)((etot + 255) / 256);

  // layer 1
  k_init  <<<nhBlocks, 256, 0, stream>>>(deg, bufB, out, N);
  k_deg   <<<(E + 255) / 256, 256, 0, stream>>>(dst, deg, E);
  k_rsqrt <<<(N + 255) / 256, 256, 0, stream>>>(deg, N);
  k_gemm_scale<<<gBlocks, 256, 0, stream>>>(x, W1, deg, bufA, N);
  k_edge  <<<eBlocks, 256, 0, stream>>>(src, dst, bufA, bufB, E);
  k_post_relu<<<nhBlocks, 256, 0, stream>>>(bufA, bufB, deg, b1, N);

  // layer 2 (GEMM in place over bufA; bufB was re-zeroed by k_post_relu)
  k_gemm_scale<<<gBlocks, 256, 0, stream>>>(bufA, W2, deg, bufA, N);
  k_edge  <<<eBlocks, 256, 0, stream>>>(src, dst, bufA, bufB, E);
  k_final <<<512, HID, 0, stream>>>(bufA, bufB, deg, b2, out, N);
}